// DiseaseOccHetGNN_20804821582167
// MI455X (gfx1250) — compile-verified
//
#include <hip/hip_runtime.h>
#include <math.h>

// ---------------------------------------------------------------------------
// DiseaseOccHetGNN for MI455X (gfx1250): 2-layer hetero GAT + LN + final linear
// fp32 GEMMs via V_WMMA_F32_16X16X4_F32 (wave32), W tile staged in LDS.
// ---------------------------------------------------------------------------

#define DD        128
#define HH        4
#define NV        15000
#define NO        40000
#define EPSLN     1e-5f
#define NEGSLOPE  0.2f
#define WPITCH    80    // float2 per K-pair row: stride 160 dwords == 32 mod 64 banks

typedef float v2f __attribute__((ext_vector_type(2)));
typedef float v8f __attribute__((ext_vector_type(8)));

// ---------------- GEMM: Y[N,C] = X[N,128] @ W[128,C] (+resid)(+bias) --------
// Block: 8 waves -> 128 rows x 64 cols. W tile (128x64) staged in LDS packed
// as K-pairs so each B fragment is one ds_load_b64. A fragments preloaded to
// registers (32 x float2 per lane). Inner loop: 128 x (ds_load_b64 + WMMA).
// A layout (ISA 7.12.2, 32-bit A 16x4): lanes 0-15 hold K0,K1; lanes 16-31 K2,K3.
// C/D layout: VGPR r -> rows r (lanes 0-15) and r+8 (lanes 16-31).
__global__ void __launch_bounds__(256)
gemm_wmma_f32(const float* __restrict__ X, const float* __restrict__ W,
              float* __restrict__ Y, int N, int C,
              const float* __restrict__ resid, const float* __restrict__ bias)
{
    __shared__ v2f sW[64 * WPITCH];   // 40 KB

    const int tid     = threadIdx.x;
    const int lane    = tid & 31;
    const int wave    = tid >> 5;
    const int rowBase = (blockIdx.x * 8 + wave) * 16;
    const int colBase = blockIdx.y * 64;

    // ---- cooperative stage of W column tile (K=128 x 64 cols), K-pair packed
    for (int i = tid; i < 64 * 64; i += 256) {
        int p = i >> 6;           // K-pair index 0..63
        int n = i & 63;           // column within tile
        v2f e;
        e.x = W[(size_t)(2 * p)     * C + colBase + n];
        e.y = W[(size_t)(2 * p + 1) * C + colBase + n];
        sW[p * WPITCH + n] = e;
    }

    // ---- preload this wave's A fragments (16 rows x K=128) into registers
    const int mRow  = rowBase + (lane & 15);
    const int mLoad = mRow < N ? mRow : (N - 1);
    const int hsel  = lane >> 4;              // 0: K%4 in {0,1}; 1: {2,3}
    const float* xrow = X + (size_t)mLoad * DD;
    v2f areg[32];
#pragma unroll
    for (int i = 0; i < 32; ++i) {
        areg[i].x = xrow[4 * i + 2 * hsel];
        areg[i].y = xrow[4 * i + 2 * hsel + 1];
    }

    __syncthreads();

    v8f acc[4] = {};
    const int nLocal = lane & 15;
#pragma unroll
    for (int i = 0; i < 32; ++i) {            // k0 = 4*i
        const v2f a = areg[i];
        const v2f* wp = sW + (2 * i + hsel) * WPITCH + nLocal;
#pragma unroll
        for (int t = 0; t < 4; ++t) {
            v2f b = wp[16 * t];
            acc[t] = __builtin_amdgcn_wmma_f32_16x16x4_f32(
                false, a, false, b, (short)0, acc[t], false, false);
        }
    }

    const int mHi = (lane >> 4) * 8;
#pragma unroll
    for (int t = 0; t < 4; ++t) {
        const int n = colBase + 16 * t + (lane & 15);
#pragma unroll
        for (int r = 0; r < 8; ++r) {
            const int mm = rowBase + r + mHi;
            if (mm < N) {
                float v = acc[t][r];
                if (bias)  v += bias[n];
                if (resid) v += resid[(size_t)mm * C + n];
                Y[(size_t)mm * C + n] = v;
            }
        }
    }
}

// ---------------- a[n,h] = sum_d h[n, h*128+d] * att[h,d] -------------------
__global__ void att_logits(const float* __restrict__ h, const float* __restrict__ att,
                           float* __restrict__ out, int N)
{
    int i = blockIdx.x * blockDim.x + threadIdx.x;
    if (i >= N * HH) return;
    int n = i >> 2, hh = i & 3;
    const float* hr = h + (size_t)n * (HH * DD) + hh * DD;
    const float* ar = att + hh * DD;
    float s = 0.f;
#pragma unroll 4
    for (int d = 0; d < DD; ++d) s += hr[d] * ar[d];
    out[i] = s;
}

// Monotone float -> uint key so atomicMax(uint) gives float max.
__device__ __forceinline__ unsigned fkey(float x) {
    unsigned b = __float_as_uint(x);
    return (b & 0x80000000u) ? ~b : (b | 0x80000000u);
}

// ---------------- per-(edge,head) leaky-relu logit + segment max ------------
__global__ void edge_logit(const float* __restrict__ a_s, const float* __restrict__ a_d,
                           const int* __restrict__ src, const int* __restrict__ dst,
                           float* __restrict__ ebuf, unsigned* __restrict__ mkey, int E)
{
    int i = blockIdx.x * blockDim.x + threadIdx.x;
    if (i >= E * HH) return;
    int e = i >> 2, hh = i & 3;
    float v = a_s[src[e] * HH + hh] + a_d[dst[e] * HH + hh];
    v = (v > 0.f) ? v : NEGSLOPE * v;
    ebuf[i] = v;
    atomicMax(mkey + dst[e] * HH + hh, fkey(v));
}

// Decode key -> float in place; empty segments (-NaN from key 0) -> 0.0
__global__ void decode_max(unsigned* __restrict__ mk, int n)
{
    int i = blockIdx.x * blockDim.x + threadIdx.x;
    if (i >= n) return;
    unsigned k = mk[i];
    unsigned b = (k & 0x80000000u) ? (k ^ 0x80000000u) : ~k;
    float v = __uint_as_float(b);
    if (!(v == v) || fabsf(v) > 3.0e38f) v = 0.f;   // !isfinite
    ((float*)mk)[i] = v;
}

// ---------------- ex = exp(e - m[dst]); den[dst] += ex ----------------------
__global__ void edge_exp(float* __restrict__ ebuf, const int* __restrict__ dst,
                         const float* __restrict__ m, float* __restrict__ den, int E)
{
    int i = blockIdx.x * blockDim.x + threadIdx.x;
    if (i >= E * HH) return;
    int e = i >> 2, hh = i & 3;
    float ex = __expf(ebuf[i] - m[dst[e] * HH + hh]);
    ebuf[i] = ex;
    atomicAdd(den + dst[e] * HH + hh, ex);
}

// ---------------- agg[dst,h,:] += alpha * hs[src,h,:] -----------------------
// One wave per (edge,head); each lane handles 4 contiguous d's. hs buffers
// (30-80MB) stay resident in the 192MB L2, so the 512MB of gather traffic
// per relation never hits HBM.
__global__ void edge_agg(const float* __restrict__ hs, const float* __restrict__ ebuf,
                         const float* __restrict__ den, const int* __restrict__ src,
                         const int* __restrict__ dst, float* __restrict__ agg, int E)
{
    int gt   = blockIdx.x * blockDim.x + threadIdx.x;
    int wid  = gt >> 5;
    int lane = threadIdx.x & 31;
    if (wid >= E * HH) return;
    int e = wid >> 2, hh = wid & 3;
    int s = src[e], d = dst[e];
    float alpha = ebuf[wid] / den[d * HH + hh];
    const float4* hr = (const float4*)(hs + (size_t)s * (HH * DD) + hh * DD);
    float4 v = hr[lane];
    float* o = agg + (size_t)d * (HH * DD) + hh * DD + lane * 4;
    atomicAdd(o + 0, v.x * alpha);
    atomicAdd(o + 1, v.y * alpha);
    atomicAdd(o + 2, v.z * alpha);
    atomicAdd(o + 3, v.w * alpha);
}

// ---------------- out[n,d] = mean_h agg[n,h,d] + bias[d] (optional +=) ------
__global__ void head_mean_bias(const float* __restrict__ agg, const float* __restrict__ bias,
                               float* __restrict__ out, int N, int accum)
{
    int i = blockIdx.x * blockDim.x + threadIdx.x;
    if (i >= N * DD) return;
    int n = i >> 7, d = i & 127;
    const float* a = agg + (size_t)n * (HH * DD) + d;
    float s = 0.25f * (a[0] + a[DD] + a[2 * DD] + a[3 * DD]) + bias[d];
    out[i] = accum ? (out[i] + s) : s;
}

// ---------------- y = LN(x + alpha*h) * g + b  (one wave per row) -----------
__global__ void add_norm(const float* __restrict__ x, const float* __restrict__ h,
                         const float* __restrict__ alphas, int aidx,
                         const float* __restrict__ g, const float* __restrict__ b,
                         float* __restrict__ y, int N)
{
    int row  = (blockIdx.x * blockDim.x + threadIdx.x) >> 5;
    int lane = threadIdx.x & 31;
    if (row >= N) return;
    float al = alphas[aidx];
    float t[4], s = 0.f, s2 = 0.f;
#pragma unroll
    for (int j = 0; j < 4; ++j) {
        int d = lane + 32 * j;
        t[j] = x[(size_t)row * DD + d] + al * h[(size_t)row * DD + d];
        s  += t[j];
        s2 += t[j] * t[j];
    }
#pragma unroll
    for (int o = 16; o > 0; o >>= 1) {
        s  += __shfl_xor(s, o, 32);
        s2 += __shfl_xor(s2, o, 32);
    }
    float mu  = s * (1.f / 128.f);
    float var = s2 * (1.f / 128.f) - mu * mu;
    float rs  = rsqrtf(var + EPSLN);
#pragma unroll
    for (int j = 0; j < 4; ++j) {
        int d = lane + 32 * j;
        y[(size_t)row * DD + d] = (t[j] - mu) * rs * g[d] + b[d];
    }
}

// ---------------------------------------------------------------------------
struct GatWs {
    float* hA;      // [NO,512] source projections
    float* hB;      // [NO,512] dest projections
    float* as_;     // [NO,H]
    float* ad_;     // [NO,H]
    unsigned* mkey; // [NO,H] max keys -> floats
    float* den;     // [NO,H]
    float* ebuf;    // [E_CO,H]
    float* agg;     // [NO,512]
};

static void run_gat(hipStream_t stream, const float* xs, int ns, const float* xd, int nd,
                    const int* src, const int* dst, int E,
                    const float* Wr, const float* attS, const float* attD, const float* br,
                    float* out, int accum, const GatWs& w, bool selfSame)
{
    dim3 blk(256);
    gemm_wmma_f32<<<dim3((ns + 127) / 128, 8), blk, 0, stream>>>(xs, Wr, w.hA, ns, 512,
                                                                 nullptr, nullptr);
    const float* hd = w.hA;
    if (!selfSame) {
        gemm_wmma_f32<<<dim3((nd + 127) / 128, 8), blk, 0, stream>>>(xd, Wr, w.hB, nd, 512,
                                                                     nullptr, nullptr);
        hd = w.hB;
    }
    att_logits<<<(ns * HH + 255) / 256, 256, 0, stream>>>(w.hA, attS, w.as_, ns);
    att_logits<<<(nd * HH + 255) / 256, 256, 0, stream>>>(hd, attD, w.ad_, nd);

    hipMemsetAsync(w.mkey, 0, (size_t)nd * HH * 4, stream);
    hipMemsetAsync(w.den, 0, (size_t)nd * HH * 4, stream);
    hipMemsetAsync(w.agg, 0, (size_t)nd * HH * DD * 4, stream);

    int EH = E * HH;
    edge_logit<<<(EH + 255) / 256, 256, 0, stream>>>(w.as_, w.ad_, src, dst, w.ebuf, w.mkey, E);
    decode_max<<<(nd * HH + 255) / 256, 256, 0, stream>>>(w.mkey, nd * HH);
    edge_exp<<<(EH + 255) / 256, 256, 0, stream>>>(w.ebuf, dst, (const float*)w.mkey, w.den, E);
    edge_agg<<<(int)(((size_t)EH * 32 + 255) / 256), 256, 0, stream>>>(w.hA, w.ebuf, w.den,
                                                                       src, dst, w.agg, E);
    head_mean_bias<<<(nd * DD + 255) / 256, 256, 0, stream>>>(w.agg, br, out, nd, accum);
}

extern "C" void kernel_launch(void* const* d_in, const int* in_sizes, int n_in,
                              void* d_out, int out_size, void* d_ws, size_t ws_size,
                              hipStream_t stream)
{
    const float* x_v     = (const float*)d_in[0];
    const float* x_o     = (const float*)d_in[1];
    const float* W1      = (const float*)d_in[2];   // [3,128,512]
    const float* as1     = (const float*)d_in[3];   // [3,4,128]
    const float* ad1     = (const float*)d_in[4];
    const float* b1      = (const float*)d_in[5];   // [3,128]
    const float* W2      = (const float*)d_in[6];
    const float* as2     = (const float*)d_in[7];
    const float* ad2     = (const float*)d_in[8];
    const float* b2      = (const float*)d_in[9];
    const float* alphas  = (const float*)d_in[10];  // [4]
    const float* ln_g    = (const float*)d_in[11];  // [4,128]
    const float* ln_b    = (const float*)d_in[12];
    const float* lin_v_w = (const float*)d_in[13];  // [128,128]
    const float* lin_v_b = (const float*)d_in[14];
    const float* lin_o_w = (const float*)d_in[15];
    const float* lin_o_b = (const float*)d_in[16];
    const int E_co = in_sizes[17] / 2;
    const int E_nx = in_sizes[19] / 2;
    const int* co = (const int*)d_in[17];
    const int* cb = (const int*)d_in[18];
    const int* nx = (const int*)d_in[19];
    const int* co_src = co;          const int* co_dst = co + E_co;
    const int* cb_src = cb;          const int* cb_dst = cb + E_co;
    const int* nx_src = nx;          const int* nx_dst = nx + E_nx;

    // Carve workspace (256B aligned chunks).
    char* p = (char*)d_ws;
    auto take = [&](size_t bytes) -> void* {
        void* q = (void*)p;
        p += (bytes + 255) & ~(size_t)255;
        return q;
    };
    GatWs w;
    w.hA   = (float*)take((size_t)NO * 512 * 4);
    w.hB   = (float*)take((size_t)NO * 512 * 4);
    w.agg  = (float*)take((size_t)NO * 512 * 4);
    w.as_  = (float*)take((size_t)NO * HH * 4);
    w.ad_  = (float*)take((size_t)NO * HH * 4);
    w.mkey = (unsigned*)take((size_t)NO * HH * 4);
    w.den  = (float*)take((size_t)NO * HH * 4);
    w.ebuf = (float*)take((size_t)E_co * HH * 4);
    float* hv = (float*)take((size_t)NV * DD * 4);
    float* ho = (float*)take((size_t)NO * DD * 4);
    float* v1 = (float*)take((size_t)NV * DD * 4);
    float* o1 = (float*)take((size_t)NO * DD * 4);
    float* v2 = (float*)take((size_t)NV * DD * 4);
    float* o2 = (float*)take((size_t)NO * DD * 4);

    const size_t WSTRIDE = (size_t)DD * 512;  // per-relation W
    const size_t ASTRIDE = (size_t)HH * DD;   // per-relation att

    // ---- Layer 1 ----
    run_gat(stream, x_v, NV, x_o, NO, co_src, co_dst, E_co,
            W1 + 0 * WSTRIDE, as1 + 0 * ASTRIDE, ad1 + 0 * ASTRIDE, b1 + 0 * DD, ho, 0, w, false);
    run_gat(stream, x_o, NO, x_v, NV, cb_src, cb_dst, E_co,
            W1 + 1 * WSTRIDE, as1 + 1 * ASTRIDE, ad1 + 1 * ASTRIDE, b1 + 1 * DD, hv, 0, w, false);
    run_gat(stream, x_v, NV, x_v, NV, nx_src, nx_dst, E_nx,
            W1 + 2 * WSTRIDE, as1 + 2 * ASTRIDE, ad1 + 2 * ASTRIDE, b1 + 2 * DD, hv, 1, w, true);
    add_norm<<<(NV * 32 + 255) / 256, 256, 0, stream>>>(x_v, hv, alphas, 0,
                                                        ln_g + 0 * DD, ln_b + 0 * DD, v1, NV);
    add_norm<<<(NO * 32 + 255) / 256, 256, 0, stream>>>(x_o, ho, alphas, 1,
                                                        ln_g + 1 * DD, ln_b + 1 * DD, o1, NO);

    // ---- Layer 2 ----
    run_gat(stream, v1, NV, o1, NO, co_src, co_dst, E_co,
            W2 + 0 * WSTRIDE, as2 + 0 * ASTRIDE, ad2 + 0 * ASTRIDE, b2 + 0 * DD, ho, 0, w, false);
    run_gat(stream, o1, NO, v1, NV, cb_src, cb_dst, E_co,
            W2 + 1 * WSTRIDE, as2 + 1 * ASTRIDE, ad2 + 1 * ASTRIDE, b2 + 1 * DD, hv, 0, w, false);
    run_gat(stream, v1, NV, v1, NV, nx_src, nx_dst, E_nx,
            W2 + 2 * WSTRIDE, as2 + 2 * ASTRIDE, ad2 + 2 * ASTRIDE, b2 + 2 * DD, hv, 1, w, true);
    add_norm<<<(NV * 32 + 255) / 256, 256, 0, stream>>>(v1, hv, alphas, 2,
                                                        ln_g + 2 * DD, ln_b + 2 * DD, v2, NV);
    add_norm<<<(NO * 32 + 255) / 256, 256, 0, stream>>>(o1, ho, alphas, 3,
                                                        ln_g + 3 * DD, ln_b + 3 * DD, o2, NO);

    // ---- Final: out = x + x @ W_lin + b_lin ----
    float* out_v = (float*)d_out;
    float* out_o = out_v + (size_t)NV * DD;
    gemm_wmma_f32<<<dim3((NV + 127) / 128, 2), 256, 0, stream>>>(v2, lin_v_w, out_v, NV, DD,
                                                                 v2, lin_v_b);
    gemm_wmma_f32<<<dim3((NO + 127) / 128, 2), 256, 0, stream>>>(o2, lin_o_w, out_o, NO, DD,
                                                                 o2, lin_o_b);
}